// ChaosModulator_28922309771717
// MI455X (gfx1250) — compile-verified
//
#include <hip/hip_runtime.h>
#include <hip/hip_bf16.h>

// Problem: z_{t+1} = clip(0.5 z + 0.25 (1 + tanh(3.5 z (1-z) + 0.5 x_t)), 0, 1)
//          u_t     = 0.5 x_t + (z_{t+1} - 0.5)
// (b,c,t) = (32,512,4096); 16384 independent serial scans of length 4096.
// HBM-bound: ~512MB traffic -> ~22us roofline at 23.3 TB/s.
//
// CDNA5 path: double-buffered global->LDS async B128 loads (coalesced tiles),
// s_wait_asynccnt pipelining, ds_load_b128 tile reads, v_tanh_f32 gate,
// nontemporal b128 output stores.

typedef float v4f __attribute__((ext_vector_type(4)));   // native 128-bit vector

#define T_LEN      4096
#define CH_PER_BLK 64            // channels (threads) per block
#define CHUNK      64            // timesteps staged per tile
#define NCHUNK     (T_LEN / CHUNK)
#define ROW_F      68            // floats per LDS row: 64 + 4 pad (16B-aligned rows)
#define TILE_F     (CH_PER_BLK * ROW_F)

// Issue one tile (64 ch x 64 t = 16KB) of async global->LDS B128 loads.
// Segment m = k*64 + tid -> row = k*4 + (tid>>4), 16B-col = tid&15.
// GVS form: saddr = block slab base (SGPR pair), vaddr = 32-bit byte offset.
__device__ __forceinline__ void issue_chunk_loads(const float* __restrict__ xblk,
                                                  uint32_t lds_buf_base,
                                                  int tid, int chunk)
{
    uint32_t voff = ((uint32_t)(tid >> 4) * T_LEN
                   + (uint32_t)(tid & 15) * 4
                   + (uint32_t)chunk * CHUNK) * 4u;          // global byte offset
    uint32_t lo   = lds_buf_base
                  + (uint32_t)(tid >> 4) * (ROW_F * 4)
                  + (uint32_t)(tid & 15) * 16;               // LDS byte address
#pragma unroll
    for (int k = 0; k < 16; ++k) {
        asm volatile("global_load_async_to_lds_b128 %0, %1, %2"
                     :: "v"(lo), "v"(voff), "s"(xblk)
                     : "memory");
        voff += 4u * T_LEN * 4u;   // +4 rows in global (65536 B)
        lo   += 4u * ROW_F * 4u;   // +4 rows in LDS    (1088 B)
    }
}

__global__ __launch_bounds__(CH_PER_BLK)
void ChaosModulator_28922309771717_kernel(const float* __restrict__ x,
                                          const float* __restrict__ z0,
                                          float* __restrict__ u)
{
    __shared__ __align__(16) float tile[2 * TILE_F];

    const int tid = threadIdx.x;
    const int ch0 = blockIdx.x * CH_PER_BLK;
    const int ch  = ch0 + tid;

    const float* xblk = x + (size_t)ch0 * T_LEN;   // block's channel slab
    float*       uout = u + (size_t)ch  * T_LEN;   // this thread's output row

    // Escapes tile via ptrtoint -> asm may write it; keeps ds loads honest.
    const uint32_t lds0 = (uint32_t)(uintptr_t)&tile[0];

    float z = z0[ch];

    // Prime buffer 0.
    issue_chunk_loads(xblk, lds0, tid, 0);

    for (int c = 0; c < NCHUNK; ++c) {
        // Prefetch next chunk into the other buffer, then wait for current.
        if (c + 1 < NCHUNK) {
            issue_chunk_loads(xblk, lds0 + (uint32_t)((c + 1) & 1) * (TILE_F * 4),
                              tid, c + 1);
            asm volatile("s_wait_asynccnt 16" ::: "memory"); // chunk c done, c+1 in flight
        } else {
            asm volatile("s_wait_asynccnt 0" ::: "memory");
        }
        __syncthreads();  // chunk c visible to all waves

        const int rbase = (c & 1) * TILE_F + tid * ROW_F;   // this thread's row
        const int t0    = c * CHUNK;
        const v4f* trow = reinterpret_cast<const v4f*>(&tile[rbase]); // ds_load_b128

#pragma unroll
        for (int j4 = 0; j4 < CHUNK / 4; ++j4) {
            const v4f xv4 = trow[j4];
            v4f uv;
#pragma unroll
            for (int q = 0; q < 4; ++q) {
                const float xv = xv4[q];
                // sigma = 3.5*z*(1-z) + 0.5*x
                const float s = fmaf(3.5f * z, 1.0f - z, 0.5f * xv);
#if __has_builtin(__builtin_amdgcn_tanhf)
                const float th = __builtin_amdgcn_tanhf(s);   // v_tanh_f32
#else
                const float th = tanhf(s);
#endif
                // z = 0.5*z + 0.25 + 0.25*tanh(s), clipped to [0,1]
                z = fmaf(0.25f, th, fmaf(0.5f, z, 0.25f));
                z = fminf(fmaxf(z, 0.0f), 1.0f);
                // u = 0.5*x + (z - 0.5)
                uv[q] = fmaf(0.5f, xv, z - 0.5f);
            }
            // Streaming store: single-use output, keep it non-temporal.
            __builtin_nontemporal_store(uv,
                reinterpret_cast<v4f*>(uout + t0 + j4 * 4));
        }
        __syncthreads();  // buffer (c&1) free for reuse by chunk c+2
    }
}

extern "C" void kernel_launch(void* const* d_in, const int* in_sizes, int n_in,
                              void* d_out, int out_size, void* d_ws, size_t ws_size,
                              hipStream_t stream) {
    const float* x  = (const float*)d_in[0];   // (32, 512, 4096) f32
    const float* z0 = (const float*)d_in[1];   // (32, 512) f32
    float* u        = (float*)d_out;           // (32, 512, 4096) f32

    const int n_channels = in_sizes[1];        // 16384
    const int n_blocks   = n_channels / CH_PER_BLK;  // 256

    ChaosModulator_28922309771717_kernel<<<n_blocks, CH_PER_BLK, 0, stream>>>(x, z0, u);
}